// bbIoU_65781719106144
// MI455X (gfx1250) — compile-verified
//
#include <hip/hip_runtime.h>

// Problem constants (B, C=1, X, Y, Z from the reference)
#define BATCHES 8
#define NX 128
#define NY 128
#define NZ 128
#define ELEMS_PER_BATCH (NX * NY * NZ)            // 2,097,152
#define CHUNKS_PER_BATCH (ELEMS_PER_BATCH / 4)    // 524,288 float4 chunks
#define BLOCKS_PER_BATCH 256
#define CHUNKS_PER_BLOCK (CHUNKS_PER_BATCH / BLOCKS_PER_BATCH)  // 2048
#define TPB 256
#define STAGES (CHUNKS_PER_BLOCK / TPB)           // 8 double-buffered stages

// 16-byte int vector types matching the async-builtin's signature
typedef int v4i __attribute__((vector_size(16)));
typedef __attribute__((address_space(1))) v4i g_v4i;   // global
typedef __attribute__((address_space(3))) v4i l_v4i;   // LDS

// ---- CDNA5 async global->LDS copy (ASYNCcnt-tracked) ------------------------
__device__ __forceinline__ void async_copy_b128(const float* gsrc, float* ldst) {
#if __has_builtin(__builtin_amdgcn_global_load_async_to_lds_b128)
  __builtin_amdgcn_global_load_async_to_lds_b128(
      (g_v4i*)gsrc, (l_v4i*)ldst, /*offset=*/0, /*cpol=*/0);
#else
  unsigned lds_off = (unsigned)(unsigned long long)(l_v4i*)ldst;
  asm volatile("global_load_async_to_lds_b128 %0, %1, off"
               :: "v"(lds_off), "v"(gsrc) : "memory");
#endif
}

#define WAIT_ASYNC(N) asm volatile("s_wait_asynccnt " #N ::: "memory")
#define WAIT_DS0()    asm volatile("s_wait_dscnt 0x0" ::: "memory")

// ---- wave32 min/max reductions via ds_swizzle (xor butterfly) ---------------
__device__ __forceinline__ int imin_i(int a, int b) { return a < b ? a : b; }
__device__ __forceinline__ int imax_i(int a, int b) { return a > b ? a : b; }

#define SWZ(v, m) __builtin_amdgcn_ds_swizzle((v), ((m) << 10) | 0x1f)
__device__ __forceinline__ int wave_min(int v) {
  v = imin_i(v, SWZ(v, 16));
  v = imin_i(v, SWZ(v, 8));
  v = imin_i(v, SWZ(v, 4));
  v = imin_i(v, SWZ(v, 2));
  v = imin_i(v, SWZ(v, 1));
  return v;
}
__device__ __forceinline__ int wave_max(int v) {
  v = imax_i(v, SWZ(v, 16));
  v = imax_i(v, SWZ(v, 8));
  v = imax_i(v, SWZ(v, 4));
  v = imax_i(v, SWZ(v, 2));
  v = imax_i(v, SWZ(v, 1));
  return v;
}

// ---- kernel 1: init workspace ----------------------------------------------
// ws layout: int ws[8][12]; [b][0..2]=pred mins (x,y,z), [3..5]=pred maxs,
//                           [6..8]=targ mins,           [9..11]=targ maxs
__global__ void bbox_init(int* __restrict__ ws) {
  int i = threadIdx.x;
  if (i < BATCHES * 12) {
    int k = i % 12;
    bool is_min = (k < 3) || (k >= 6 && k < 9);
    ws[i] = is_min ? 128 : -1;   // matches reference fill values (size / -1)
  }
}

// ---- kernel 2: streaming bbox scan -----------------------------------------
__global__ __launch_bounds__(TPB) void bbox_scan(const float* __restrict__ pred,
                                                 const float* __restrict__ targ,
                                                 int* __restrict__ ws) {
  __shared__ __align__(16) float lp[2][TPB * 4];
  __shared__ __align__(16) float lt[2][TPB * 4];
  __shared__ int sred[12];

  const int tid = threadIdx.x;
  const int bx  = blockIdx.x;
  const int b   = blockIdx.y;
  const size_t batch_base = (size_t)b * ELEMS_PER_BATCH;
  const int chunk0 = bx * CHUNKS_PER_BLOCK;

  int pxmin = NX, pxmax = -1, pymin = NY, pymax = -1, pzmin = NZ, pzmax = -1;
  int txmin = NX, txmax = -1, tymin = NY, tymax = -1, tzmin = NZ, tzmax = -1;

  // prime the pipeline: stage 0 into buffer 0 (per-wave buffers; each lane
  // reads back exactly the 16B it copied, so no block barrier is needed)
  {
    int c = chunk0 + tid;
    async_copy_b128(pred + batch_base + (size_t)c * 4, &lp[0][tid * 4]);
    async_copy_b128(targ + batch_base + (size_t)c * 4, &lt[0][tid * 4]);
  }

  for (int it = 0; it < STAGES; ++it) {
    if (it + 1 < STAGES) {
      // ensure previous ds reads of the buffer we're about to overwrite are done
      WAIT_DS0();
      int c = chunk0 + (it + 1) * TPB + tid;
      int nbuf = (it + 1) & 1;
      async_copy_b128(pred + batch_base + (size_t)c * 4, &lp[nbuf][tid * 4]);
      async_copy_b128(targ + batch_base + (size_t)c * 4, &lt[nbuf][tid * 4]);
      WAIT_ASYNC(2);   // current stage's 2 copies complete; next stage in flight
    } else {
      WAIT_ASYNC(0);
    }

    int buf = it & 1;
    float4 p = *(const float4*)&lp[buf][tid * 4];
    float4 t = *(const float4*)&lt[buf][tid * 4];

    int c  = chunk0 + it * TPB + tid;
    int x  = c >> 12;            // c / 4096
    int y  = (c >> 5) & 127;     // (c / 32) % 128
    int zb = (c & 31) << 2;      // 4 * (c % 32)

    // sigmoid(p) > 0.5  <=>  p > 0
    int mp = (p.x > 0.f) | ((p.y > 0.f) << 1) | ((p.z > 0.f) << 2) | ((p.w > 0.f) << 3);
    if (mp) {
      pxmin = imin_i(pxmin, x);  pxmax = imax_i(pxmax, x);
      pymin = imin_i(pymin, y);  pymax = imax_i(pymax, y);
      pzmin = imin_i(pzmin, zb + (__ffs(mp) - 1));
      pzmax = imax_i(pzmax, zb + (31 - __clz(mp)));
    }
    int mt = (t.x > 0.5f) | ((t.y > 0.5f) << 1) | ((t.z > 0.5f) << 2) | ((t.w > 0.5f) << 3);
    if (mt) {
      txmin = imin_i(txmin, x);  txmax = imax_i(txmax, x);
      tymin = imin_i(tymin, y);  tymax = imax_i(tymax, y);
      tzmin = imin_i(tzmin, zb + (__ffs(mt) - 1));
      tzmax = imax_i(tzmax, zb + (31 - __clz(mt)));
    }
  }

  // wave32 butterfly reduction
  pxmin = wave_min(pxmin); pxmax = wave_max(pxmax);
  pymin = wave_min(pymin); pymax = wave_max(pymax);
  pzmin = wave_min(pzmin); pzmax = wave_max(pzmax);
  txmin = wave_min(txmin); txmax = wave_max(txmax);
  tymin = wave_min(tymin); tymax = wave_max(tymax);
  tzmin = wave_min(tzmin); tzmax = wave_max(tzmax);

  if (tid == 0) {
    for (int k = 0; k < 12; ++k) {
      bool is_min = (k < 3) || (k >= 6 && k < 9);
      sred[k] = is_min ? 128 : -1;
    }
  }
  __syncthreads();

  if ((tid & 31) == 0) {   // one lane per wave -> LDS atomics
    atomicMin(&sred[0], pxmin); atomicMin(&sred[1], pymin); atomicMin(&sred[2], pzmin);
    atomicMax(&sred[3], pxmax); atomicMax(&sred[4], pymax); atomicMax(&sred[5], pzmax);
    atomicMin(&sred[6], txmin); atomicMin(&sred[7], tymin); atomicMin(&sred[8], tzmin);
    atomicMax(&sred[9], txmax); atomicMax(&sred[10], tymax); atomicMax(&sred[11], tzmax);
  }
  __syncthreads();

  if (tid == 0) {          // one thread per block -> global atomics
    int* w = ws + b * 12;
    atomicMin(&w[0], sred[0]);  atomicMin(&w[1], sred[1]);  atomicMin(&w[2], sred[2]);
    atomicMax(&w[3], sred[3]);  atomicMax(&w[4], sred[4]);  atomicMax(&w[5], sred[5]);
    atomicMin(&w[6], sred[6]);  atomicMin(&w[7], sred[7]);  atomicMin(&w[8], sred[8]);
    atomicMax(&w[9], sred[9]);  atomicMax(&w[10], sred[10]); atomicMax(&w[11], sred[11]);
  }
}

// ---- kernel 3: IoU + mean ---------------------------------------------------
__global__ void bbox_finalize(const int* __restrict__ ws, float* __restrict__ out) {
  if (threadIdx.x != 0 || blockIdx.x != 0) return;
  float acc = 0.f;
  for (int b = 0; b < BATCHES; ++b) {
    const int* w = ws + b * 12;
    bool hp = w[3] >= 0;      // any pred voxel set
    bool ht = w[9] >= 0;      // any target voxel set
    float pb[6], tb[6];
    for (int k = 0; k < 6; ++k) pb[k] = hp ? (float)w[k]     : 0.f;
    for (int k = 0; k < 6; ++k) tb[k] = ht ? (float)w[6 + k] : 0.f;
    float inter = 1.f, v1 = 1.f, v2 = 1.f;
    for (int i = 0; i < 3; ++i) {
      float lo = fmaxf(pb[i], tb[i]);
      float hi = fminf(pb[3 + i], tb[3 + i]);
      inter *= fmaxf(hi - lo, 0.f);
      v1    *= (pb[3 + i] - pb[i]);
      v2    *= (tb[3 + i] - tb[i]);
    }
    float un = v1 + v2 - inter;
    acc += (un > 0.f) ? (inter / un) : 0.f;
  }
  *out = acc * 0.125f;   // mean over 8 batches
}

extern "C" void kernel_launch(void* const* d_in, const int* in_sizes, int n_in,
                              void* d_out, int out_size, void* d_ws, size_t ws_size,
                              hipStream_t stream) {
  const float* pred = (const float*)d_in[0];
  const float* targ = (const float*)d_in[1];
  int*   ws  = (int*)d_ws;
  float* out = (float*)d_out;

  bbox_init<<<dim3(1), dim3(128), 0, stream>>>(ws);
  bbox_scan<<<dim3(BLOCKS_PER_BATCH, BATCHES), dim3(TPB), 0, stream>>>(pred, targ, ws);
  bbox_finalize<<<dim3(1), dim3(32), 0, stream>>>(ws, out);
}